// SOMSoftLayer_73521250173030
// MI455X (gfx1250) — compile-verified
//
#include <hip/hip_runtime.h>

// SOM activation_response on MI455X (gfx1250, wave32).
//
// Roofline: output one-hot = 410 MB stores -> ~18us at 23.3 TB/s; this is the
// floor. GEMM is only 26 GFLOP, done exactly in f32 on the matrix pipe via
// V_WMMA_F32_16X16X4_F32 so argmin matches the f32 reference rounding-wise.
// argmin d2 == argmax (x.w - 0.5*|w|^2): no [N,K] score matrix materialized.

typedef float v2f __attribute__((ext_vector_type(2)));
typedef float v8f __attribute__((ext_vector_type(8)));

#define SOM_DIM   128
#define SOM_K     1024
#define SOM_N     100000
#define SOM_NT    (SOM_N / 16)      // 6250 sample tiles of 16
#define LDS_STRIDE 132              // 128 + 4 pad: bank = (4*row + col) % 64, conflict-free

// ---------------- kernel 0: hwsq[k] = 0.5 * sum_d w[k][d]^2 ----------------
__global__ void som_wsq_kernel(const float* __restrict__ w,
                               float* __restrict__ hwsq) {
  int k = blockIdx.x * blockDim.x + threadIdx.x;
  if (k < SOM_K) {
    const float* row = w + (size_t)k * SOM_DIM;
    float s = 0.f;
#pragma unroll 8
    for (int d = 0; d < SOM_DIM; ++d) { float v = row[d]; s += v * v; }
    hwsq[k] = 0.5f * s;
  }
}

// ---------------- kernel 1: bmu[n] = argmax_k (x.w - 0.5|w|^2) -------------
// 4 waves/block; each wave owns one 16-sample tile; w tiles staged in LDS.
__global__ __launch_bounds__(128) void som_bmu_kernel(
    const float* __restrict__ x, const float* __restrict__ w,
    const float* __restrict__ hwsq, int* __restrict__ bmu) {
  __shared__ float wt[16 * LDS_STRIDE];   // 16 protos x 128 dims, padded

  const int tid  = threadIdx.x;
  const int lane = tid & 31;
  const int wave = tid >> 5;
  const int row  = lane & 15;             // M (samples) for A, N (protos) for B/C col
  const int half = lane >> 4;
  const int off  = half * 2;              // K sub-offset per f32 16x16x4 layout

  int t = blockIdx.x * 4 + wave;
  if (t >= SOM_NT) t = SOM_NT - 1;        // clamp: duplicate work, identical writes
  const int n0 = t * 16;

  // Preload full A panel (16 samples x 128 dims) into 64 VGPRs.
  // A 16x4 f32 layout: lanes 0-15 hold (M=lane, K=0,1); lanes 16-31 (M=lane-16, K=2,3).
  v2f a[32];
#pragma unroll
  for (int s = 0; s < 32; ++s) {
    const float* p = x + (size_t)(n0 + row) * SOM_DIM + 4 * s + off;
    a[s] = *(const v2f*)p;
  }

  float best[8];
  int   bidx[8];
#pragma unroll
  for (int j = 0; j < 8; ++j) { best[j] = -3.402823466e38f; bidx[j] = 0; }

  for (int kt = 0; kt < SOM_K / 16; ++kt) {
    __syncthreads();
    // Cooperative stage of the 16x128 w tile into padded LDS (float4 per thread x4).
    {
      const float* wbase = w + (size_t)kt * 16 * SOM_DIM;
#pragma unroll
      for (int i = 0; i < 4; ++i) {
        int idx = tid * 4 + i * 512;
        int r = idx >> 7, c = idx & 127;
        *(float4*)&wt[r * LDS_STRIDE + c] = *(const float4*)&wbase[r * SOM_DIM + c];
      }
    }
    __syncthreads();

    float hw = hwsq[kt * 16 + row];       // bias for this lane's prototype column

    v8f c = {};
#pragma unroll
    for (int s = 0; s < 32; ++s) {
      // B 4x16 f32 layout mirrors A: lane holds column N=row, K pair = off,off+1.
      v2f b = *(const v2f*)&wt[row * LDS_STRIDE + 4 * s + off];
      c = __builtin_amdgcn_wmma_f32_16x16x4_f32(
          /*neg_a=*/false, a[s], /*neg_b=*/false, b,
          /*c_mod=*/(short)0, c, /*reuse_a=*/false, /*reuse_b=*/false);
    }

    const int kcol = kt * 16 + row;
#pragma unroll
    for (int j = 0; j < 8; ++j) {
      float sc = c[j] - hw;               // row M = j + 8*half, col N = kcol
      if (sc > best[j]) { best[j] = sc; bidx[j] = kcol; }  // k increases: ties keep first
    }
  }

  // Reduce over the 16 prototype columns (lanes within each half), ties -> smaller k.
#pragma unroll
  for (int j = 0; j < 8; ++j) {
    float v = best[j];
    int   idx = bidx[j];
#pragma unroll
    for (int m = 1; m < 16; m <<= 1) {
      float ov = __shfl_xor(v, m, 32);
      int   oi = __shfl_xor(idx, m, 32);
      if (ov > v || (ov == v && oi < idx)) { v = ov; idx = oi; }
    }
    if ((lane & 15) == 0) bmu[n0 + half * 8 + j] = idx;
  }
}

// ---------------- kernel 2: out[n, :] = one_hot(bmu[n]) --------------------
// Coalesced float4 stores; writes every element (no memset needed).
__global__ __launch_bounds__(256) void som_onehot_kernel(
    const int* __restrict__ bmu, float* __restrict__ out) {
  const int n = blockIdx.x;
  const int q = threadIdx.x;              // 0..255, 4 elements each
  const int b = bmu[n];
  const int k = q * 4;
  float4 v;
  v.x = (k     == b) ? 1.0f : 0.0f;
  v.y = (k + 1 == b) ? 1.0f : 0.0f;
  v.z = (k + 2 == b) ? 1.0f : 0.0f;
  v.w = (k + 3 == b) ? 1.0f : 0.0f;
  *(float4*)&out[(size_t)n * SOM_K + k] = v;
}

extern "C" void kernel_launch(void* const* d_in, const int* in_sizes, int n_in,
                              void* d_out, int out_size, void* d_ws, size_t ws_size,
                              hipStream_t stream) {
  const float* x = (const float*)d_in[0];          // [100000, 128] f32
  const float* w = (const float*)d_in[1];          // [32, 32, 128] f32 -> [1024, 128]
  float* hwsq = (float*)d_ws;                      // 1024 floats
  int*   bmu  = (int*)((char*)d_ws + 4096);        // 100000 ints
  float* out  = (float*)d_out;                     // [100000, 1024] f32

  som_wsq_kernel<<<(SOM_K + 255) / 256, 256, 0, stream>>>(w, hwsq);
  som_bmu_kernel<<<(SOM_NT + 3) / 4, 128, 0, stream>>>(x, w, hwsq, bmu);
  som_onehot_kernel<<<SOM_N, 256, 0, stream>>>(bmu, out);
}